// GatedRecNNTokenEmbedder_86483461472758
// MI455X (gfx1250) — compile-verified
//
#include <hip/hip_runtime.h>
#include <hip/hip_bf16.h>

typedef float v2f __attribute__((ext_vector_type(2)));
typedef float v8f __attribute__((ext_vector_type(8)));

#define WMMA_F32(a, b, c) \
  __builtin_amdgcn_wmma_f32_16x16x4_f32(false, (a), false, (b), (short)0, (c), false, false)

// ---------------------------------------------------------------------------
// Problem constants
// ---------------------------------------------------------------------------
#define N_TOK   8192      // B*SW = 32*256
#define SC      16
#define CDIM    128
#define WDIM    512
#define EDIM    640       // WDIM + CDIM
#define ODIM    1024

// ---------------------------------------------------------------------------
// Kernel 1: fused char RNN (+ word gather).  16 tokens / WG, 1 wave / token.
// Weight panels are double-buffered in LDS and filled with async DMA
// (global_load_async_to_lds_b128, ASYNCcnt) overlapping the WMMA phases.
// ---------------------------------------------------------------------------
#define TOK_PER_WG 16
#define LSTRIDE    132                 // padded layer row stride (floats)
#define TOK_LDS    (16 * LSTRIDE)      // 2112 floats per token
#define PSTRIDE    260                 // padded weight-panel row stride
#define PANEL_ELEMS (64 * PSTRIDE)     // 16640 floats = 66560 B per buffer

// Issue one 64-row x 256-col weight panel (d-tile t) as async loads into LDS.
// 512 threads x 8 x b128 = 64 KB.  Tracked by ASYNCcnt; no wait here.
__device__ __forceinline__
void issue_panel_async(float* sPanelBuf,
                       const float* __restrict__ Wg,
                       const float* __restrict__ Wp,
                       int t, int tid)
{
  int rowId = tid >> 3;                // 0..63  (g0,g1,g2 from Wg; z from Wp)
  int kBase = (tid & 7) * 32;          // 0..224 (floats)
  int g = rowId >> 4, j = rowId & 15;
  const float* src = (g < 3) ? (Wg + (size_t)(g * 128 + t * 16 + j) * 256)
                             : (Wp + (size_t)(t * 16 + j) * 256);
  unsigned long long ga = (unsigned long long)(uintptr_t)(src + kBase);
  unsigned int       la = (unsigned int)(uintptr_t)(sPanelBuf + rowId * PSTRIDE + kBase);
  #pragma unroll
  for (int q = 0; q < 8; ++q) {
    asm volatile("global_load_async_to_lds_b128 %0, %1, off offset:%2"
                 :: "v"(la), "v"(ga), "i"(q * 16) : "memory");
  }
}

__global__ __launch_bounds__(512)
void rnn_embed_kernel(const int* __restrict__ word_inputs,
                      const int* __restrict__ char_ids,
                      const int* __restrict__ char_lengths,
                      const float* __restrict__ word_table,
                      const float* __restrict__ char_table,
                      const float* __restrict__ Wg, const float* __restrict__ bg,
                      const float* __restrict__ Wp, const float* __restrict__ bp,
                      float* __restrict__ tokEmb)
{
  __shared__ float sLayer[TOK_PER_WG * TOK_LDS];  // 135168 B
  __shared__ float sPanel[2 * PANEL_ELEMS];       // 133120 B (double buffer)

  const int tid    = threadIdx.x;
  const int lane   = tid & 31;
  const int wave   = tid >> 5;              // 0..15
  const int wgTok0 = blockIdx.x * TOK_PER_WG;
  const int myTok  = wgTok0 + wave;
  float* L = sLayer + wave * TOK_LDS;

  // prefetch first weight panel (d-tile 0) into buffer 0, overlapping init
  issue_panel_async(sPanel, Wg, Wp, 0, tid);

  // ---- layer0 = char_table[char_ids[tok]]  (wave-private) ----
  {
    const int* cid = char_ids + (size_t)myTok * SC;
    #pragma unroll 4
    for (int c = 0; c < SC; ++c) {
      int id = cid[c];
      float4 v = ((const float4*)(char_table + (size_t)id * CDIM))[lane];
      *(float4*)(L + c * LSTRIDE + lane * 4) = v;
    }
  }

  // ---- word-embedding gather straight to tokEmb[:, 0:512] ----
  {
    // 16 tokens * 512 floats = 2048 float4, 512 threads -> 4 each
    #pragma unroll
    for (int r = 0; r < 4; ++r) {
      int idx  = tid + r * 512;
      int tok  = idx >> 7;                  // 128 float4 per token
      int part = idx & 127;
      int wid  = word_inputs[wgTok0 + tok];
      float4 v = ((const float4*)(word_table + (size_t)wid * WDIM))[part];
      ((float4*)(tokEmb + (size_t)(wgTok0 + tok) * EDIM))[part] = v;
    }
  }

  const int myLen = char_lengths[myTok];

  // per-lane WMMA fragment addressing
  const int n16 = lane & 15;                // M (A rows) / N (B cols) / col in C
  const int h   = lane >> 4;                // K-pair selector
  const float* Abase = L + n16 * LSTRIDE + 2 * h;

  for (int step = 1; step <= 15; ++step) {
    const int active = myLen - step;        // update positions [0, active)
    v8f ret[8];

    #pragma unroll
    for (int t = 0; t < 8; ++t) {
      const int cur = t & 1;
      // panel t has landed in buffer `cur` (my async ops done) ...
      asm volatile("s_wait_asynccnt 0x0" ::: "memory");
      // ... and after the barrier everyone's writes are visible; also the
      // other buffer's readers (phase t-1 compute) are all done.
      __syncthreads();
      // prefetch panel for the NEXT phase into the other buffer (async DMA
      // overlaps this phase's WMMA work). Weights repeat every step, so the
      // phase after t=7 is t=0 of the next step.
      issue_panel_async(sPanel + (cur ^ 1) * PANEL_ELEMS, Wg, Wp, (t + 1) & 7, tid);

      const float* B0 = sPanel + cur * PANEL_ELEMS + n16 * PSTRIDE + 2 * h;
      const float* B1 = B0 + 16 * PSTRIDE;
      const float* B2 = B0 + 32 * PSTRIDE;
      const float* B3 = B0 + 48 * PSTRIDE;

      // ---- GEMM: 16x16 tiles g0,g1,g2,z over K=256 ----
      v8f c0 = {0,0,0,0,0,0,0,0}, c1 = c0, c2 = c0, cz = c0;
      #pragma unroll 4
      for (int k0 = 0; k0 < 128; k0 += 4) {     // left half (k < 128)
        v2f a  = *(const v2f*)(Abase + k0);
        v2f w0 = *(const v2f*)(B0 + k0);
        v2f w1 = *(const v2f*)(B1 + k0);
        v2f w2 = *(const v2f*)(B2 + k0);
        v2f w3 = *(const v2f*)(B3 + k0);
        c0 = WMMA_F32(a, w0, c0);  c1 = WMMA_F32(a, w1, c1);
        c2 = WMMA_F32(a, w2, c2);  cz = WMMA_F32(a, w3, cz);
      }
      #pragma unroll 4
      for (int k0 = 128; k0 < 256; k0 += 4) {   // right half (k >= 128): +pad skip
        v2f a  = *(const v2f*)(Abase + k0 + (LSTRIDE - 128));
        v2f w0 = *(const v2f*)(B0 + k0);
        v2f w1 = *(const v2f*)(B1 + k0);
        v2f w2 = *(const v2f*)(B2 + k0);
        v2f w3 = *(const v2f*)(B3 + k0);
        c0 = WMMA_F32(a, w0, c0);  c1 = WMMA_F32(a, w1, c1);
        c2 = WMMA_F32(a, w2, c2);  cz = WMMA_F32(a, w3, cz);
      }

      // ---- gating: bias + 3-way softmax + convex combination (in regs) ----
      {
        int col = t * 16 + n16;
        float bg0 = bg[col], bg1 = bg[128 + col], bg2 = bg[256 + col];
        float bzv = bp[col];
        #pragma unroll
        for (int i = 0; i < 8; ++i) {
          int pos = i + 8 * h;
          int pr  = (pos < 15) ? pos + 1 : pos;          // clamp pad row
          float left  = L[pos * LSTRIDE + col];
          float right = L[pr  * LSTRIDE + col];
          float zc = cz[i] + bzv;
          float x0 = c0[i] + bg0, x1 = c1[i] + bg1, x2 = c2[i] + bg2;
          float mx = fmaxf(x0, fmaxf(x1, x2));
          float e0 = __expf(x0 - mx), e1 = __expf(x1 - mx), e2 = __expf(x2 - mx);
          float inv = 1.0f / (e0 + e1 + e2);
          ret[t][i] = (e0 * left + e1 * right + e2 * zc) * inv;
        }
      }
    }

    // ---- masked in-place layer update (wave owns its token) ----
    #pragma unroll
    for (int t = 0; t < 8; ++t) {
      #pragma unroll
      for (int i = 0; i < 8; ++i) {
        int pos = i + 8 * h;
        if (pos < active)
          L[pos * LSTRIDE + t * 16 + n16] = ret[t][i];
      }
    }
  }

  // ---- char_repr = layer[0] -> tokEmb[:, 512:640] ----
  {
    float4 v = *(const float4*)(L + lane * 4);
    *(float4*)(tokEmb + (size_t)myTok * EDIM + WDIM + lane * 4) = v;
  }
}

// ---------------------------------------------------------------------------
// Kernel 2: out = tokEmb(8192x640) @ W_out^T(640x1024) + b_out
// 128x128 block tile, K-chunks of 32, fp32 WMMA, 8 waves x (2M x 4N) tiles.
// ---------------------------------------------------------------------------
#define K2_STR 36   // padded LDS stride: 36*4 B rows keep float4 stores aligned

__global__ __launch_bounds__(256)
void out_gemm_kernel(const float* __restrict__ tokEmb,
                     const float* __restrict__ W_out,
                     const float* __restrict__ b_out,
                     float* __restrict__ out)
{
  __shared__ float As[128 * K2_STR];
  __shared__ float Bs[128 * K2_STR];

  const int tid  = threadIdx.x;
  const int lane = tid & 31;
  const int wave = tid >> 5;                // 0..7
  const int n16  = lane & 15;
  const int h    = lane >> 4;

  const int mBlock = blockIdx.y * 128;
  const int nBlock = blockIdx.x * 128;
  const int waveM  = wave & 3;              // 2 M-tiles each
  const int waveN  = wave >> 2;             // 4 N-tiles each

  v8f acc[2][4];
  #pragma unroll
  for (int a = 0; a < 2; ++a)
    #pragma unroll
    for (int b = 0; b < 4; ++b) acc[a][b] = (v8f){0,0,0,0,0,0,0,0};

  const float* Aw0 = As + (waveM * 32 + n16) * K2_STR + 2 * h;
  const float* Aw1 = Aw0 + 16 * K2_STR;
  const float* Bw  = Bs + (waveN * 64 + n16) * K2_STR + 2 * h;

  for (int kc = 0; kc < EDIM; kc += 32) {
    __syncthreads();
    // stage A (128 rows x 32 k) and B (128 cols x 32 k), float4 per thread x4
    #pragma unroll
    for (int p = 0; p < 4; ++p) {
      int idx = tid + p * 256;              // 0..1023 float4 slots
      int row = idx >> 3;
      int kq  = (idx & 7) * 4;
      float4 va = *(const float4*)(tokEmb + (size_t)(mBlock + row) * EDIM + kc + kq);
      float4 vb = *(const float4*)(W_out  + (size_t)(nBlock + row) * EDIM + kc + kq);
      *(float4*)(As + row * K2_STR + kq) = va;
      *(float4*)(Bs + row * K2_STR + kq) = vb;
    }
    __syncthreads();

    #pragma unroll
    for (int k0 = 0; k0 < 32; k0 += 4) {
      v2f a0 = *(const v2f*)(Aw0 + k0);
      v2f a1 = *(const v2f*)(Aw1 + k0);
      #pragma unroll
      for (int j = 0; j < 4; ++j) {
        v2f b = *(const v2f*)(Bw + j * 16 * K2_STR + k0);
        acc[0][j] = WMMA_F32(a0, b, acc[0][j]);
        acc[1][j] = WMMA_F32(a1, b, acc[1][j]);
      }
    }
  }

  // epilogue: + b_out, store
  #pragma unroll
  for (int mi = 0; mi < 2; ++mi) {
    #pragma unroll
    for (int j = 0; j < 4; ++j) {
      int col  = nBlock + waveN * 64 + j * 16 + n16;
      float bv = b_out[col];
      int rowB = mBlock + waveM * 32 + mi * 16 + 8 * h;
      #pragma unroll
      for (int i = 0; i < 8; ++i)
        out[(size_t)(rowB + i) * ODIM + col] = acc[mi][j][i] + bv;
    }
  }
}

// ---------------------------------------------------------------------------
extern "C" void kernel_launch(void* const* d_in, const int* in_sizes, int n_in,
                              void* d_out, int out_size, void* d_ws, size_t ws_size,
                              hipStream_t stream) {
  const int*   word_inputs  = (const int*)  d_in[0];
  const int*   char_ids     = (const int*)  d_in[1];
  const int*   char_lengths = (const int*)  d_in[2];
  const float* word_table   = (const float*)d_in[3];
  const float* char_table   = (const float*)d_in[4];
  const float* Wg           = (const float*)d_in[5];
  const float* bg           = (const float*)d_in[6];
  const float* Wp           = (const float*)d_in[7];
  const float* bp           = (const float*)d_in[8];
  const float* W_out        = (const float*)d_in[9];
  const float* b_out        = (const float*)d_in[10];
  float* tokEmb = (float*)d_ws;            // 8192*640*4 = 20.97 MB scratch
  float* out    = (float*)d_out;

  rnn_embed_kernel<<<N_TOK / TOK_PER_WG, 512, 0, stream>>>(
      word_inputs, char_ids, char_lengths, word_table, char_table,
      Wg, bg, Wp, bp, tokEmb);

  dim3 g2(ODIM / 128, N_TOK / 128);        // (8, 64)
  out_gemm_kernel<<<g2, 256, 0, stream>>>(tokEmb, W_out, b_out, out);
}